// BatteryRNNCell_3040836845678
// MI455X (gfx1250) — compile-verified
//
#include <hip/hip_runtime.h>
#include <hip/hip_bf16.h>
#include <cstdint>

// Battery RNN cell, B=4M elements, pure f32 streaming kernel.
// Roofline: ~80 B/element -> 320 MB @ 23.3 TB/s ~= 14 us; VALU is near that budget,
// so trans ops use the gfx1250 hardware paths (v_tanh_f32/v_log/v_sqrt/v_rcp/v_med3).
// CDNA5 paths: global_load_async_to_lds_b128 (ASYNCcnt, saddr/GVS form) for the [B,8]
// state rows, overlapped with regular b32 loads of i/qMax/Ro; ds_load_b128 readback
// from a bank-conflict-free LDS layout.

namespace {
constexpr int TPB = 256;   // 8 waves (wave32)
constexpr int EPT = 4;     // elements per thread -> 1024 elements per block

// Constants folded from the reference (double-precision fold, then f32).
constexpr float RF        = (float)(8.3144621 / 96487.0);        // R/F
constexpr float RF2       = (float)(2.0 * 8.3144621 / 96487.0);  // R/F/alpha (alpha=0.5)
constexpr float INV_VOLB  = (float)(1.0 / 1.98e-05);             // 1/(VOL-VOLS)
constexpr float INV_VOLS  = (float)(1.0 / 2.2e-06);
constexpr float INV_TDIFF = (float)(1.0 / 7.0e6);
constexpr float INV_TSN   = (float)(1.0 / 90.0);
} // namespace

__device__ __forceinline__ float tanh_f(float x) {
#if __has_builtin(__builtin_amdgcn_tanhf)
  return __builtin_amdgcn_tanhf(x);            // v_tanh_f32 (gfx1250 TRANS unit)
#else
  // Hardware tanh via asm; v_nop satisfies the TRANS-op result hazard (1 indep op).
  float r;
  asm("v_tanh_f32 %0, %1\n\tv_nop" : "=v"(r) : "v"(x));
  return r;
#endif
}

__device__ __forceinline__ float clamp_f(float x, float lo, float hi) {
  return __builtin_amdgcn_fmed3f(x, lo, hi);   // v_med3_f32: clamp in one op
}

__device__ __forceinline__ float asinh_f(float x) {
  float ax = fabsf(x);
  float r = __logf(ax + sqrtf(fmaf(ax, ax, 1.0f)));
  return copysignf(r, x);
}

__global__ void __launch_bounds__(TPB)
battery_cell_kernel(const float* __restrict__ inputs,
                    const float* __restrict__ states,
                    const float* __restrict__ qMax,
                    const float* __restrict__ Ro,
                    const float* __restrict__ Wp0, const float* __restrict__ bp0,
                    const float* __restrict__ Wp2, const float* __restrict__ bp2,
                    const float* __restrict__ Wp4, const float* __restrict__ bp4,
                    const float* __restrict__ Wn,  const float* __restrict__ bn,
                    float* __restrict__ out, int n) {
  const int tid  = threadIdx.x;
  const int base = (int)blockIdx.x * (TPB * EPT);   // B=4M: fits i32 easily

  // Split each 32B state row into two 16B halves in separate buffers so the
  // ds_load_b128 readback has a 16B lane stride (all 64 banks hit, no conflicts).
  __shared__ __align__(16) float sA[EPT][TPB * 4];  // Tb, Vo, Vsn, Vsp
  __shared__ __align__(16) float sB[EPT][TPB * 4];  // qnB, qnS, qpB, qpS

  int sidx[EPT];
#pragma unroll
  for (int e = 0; e < EPT; ++e) {
    const int idx = base + e * TPB + tid;
    sidx[e] = (idx < n) ? idx : (n - 1);            // clamp OOB lanes (i32 cmp)
  }

  // ---- Stage state rows via the gfx1250 async global->LDS path (ASYNCcnt) ----
  // GVS/saddr form: uniform base in SGPR pair + 32-bit per-lane byte offset
  // (max offset = 4M*32B = 128MB, fits in i32) -> no per-lane 64-bit adds.
#pragma unroll
  for (int e = 0; e < EPT; ++e) {
    const unsigned voff   = (unsigned)sidx[e] << 5;  // row byte offset
    const unsigned voff16 = voff | 16u;              // second 16B half (rows 32B-aligned)
    const unsigned la = (unsigned)(uintptr_t)(&sA[e][tid * 4]);  // low 32 bits = LDS offset
    const unsigned lb = (unsigned)(uintptr_t)(&sB[e][tid * 4]);
    asm volatile("global_load_async_to_lds_b128 %0, %1, %2"
                 :: "v"(la), "v"(voff),   "s"(states) : "memory");
    asm volatile("global_load_async_to_lds_b128 %0, %1, %2"
                 :: "v"(lb), "v"(voff16), "s"(states) : "memory");
  }

  // ---- Overlap with the async transfers: per-element scalar loads ----
  float iv[EPT], qMv[EPT], Rov[EPT];
#pragma unroll
  for (int e = 0; e < EPT; ++e) {
    iv[e]  = inputs[sidx[e]];
    qMv[e] = qMax[sidx[e]];
    Rov[e] = Ro[sidx[e]];
  }

  // ---- Uniform weights (SMEM-promoted, L2-hot; amortized over EPT elements) ----
  float w0[8], c0[8];
#pragma unroll
  for (int j = 0; j < 8; ++j) { w0[j] = Wp0[j]; c0[j] = bp0[j]; }
  float w2[32];
#pragma unroll
  for (int j = 0; j < 32; ++j) w2[j] = Wp2[j];
  float c2[4], w4[4];
#pragma unroll
  for (int k = 0; k < 4; ++k) { c2[k] = bp2[k]; w4[k] = Wp4[k]; }
  const float c4  = bp4[0];
  const float wnv = Wn[0];
  const float cnv = bn[0];

#if __has_builtin(__builtin_amdgcn_s_wait_asynccnt)
  __builtin_amdgcn_s_wait_asynccnt(0);
#else
  asm volatile("s_wait_asynccnt 0" ::: "memory");
#endif

  // ---- Per-element cell update ----
#pragma unroll
  for (int e = 0; e < EPT; ++e) {
    const int idx = base + e * TPB + tid;

    const float4 s03 = *reinterpret_cast<const float4*>(&sA[e][tid * 4]);
    const float4 s47 = *reinterpret_cast<const float4*>(&sB[e][tid * 4]);
    const float Tb = s03.x, Vo = s03.y, Vsn = s03.z, Vsp = s03.w;
    const float qnB = s47.x, qnS = s47.y, qpB = s47.z, qpS = s47.w;
    const float i_ = iv[e];

    // qSMax = qMax * QMAX_BASE * VOLS / VOL = qMax * 1000
    const float qSMax = qMv[e] * 1000.0f;
    const float rq = __builtin_amdgcn_rcpf(qSMax);

    // --- getNextState ---
    const float xpS = clamp_f(qpS * rq, 1e-18f, 1.0f);
    const float xnS = clamp_f(qnS * rq, 1e-18f, 1.0f);
    const float Jn0 = fmaf(20000.0f, sqrtf(xnS * (1.0f - xnS)), 1e-18f);
    const float Jp0 = fmaf(20000.0f, sqrtf(xpS * (1.0f - xpS)), 1e-18f);

    const float qdn = (qnB * INV_VOLB - qnS * INV_VOLS) * INV_TDIFF;
    const float qdp = (qpB * INV_VOLB - qpS * INV_VOLS) * INV_TDIFF;

    const float VoNom = i_ * Rov[e] * 10.0f;       // i * Ro * RO_BASE
    const float rTbA  = RF2 * Tb;                  // R*Tb/F/alpha
    const float halfJ = i_ * 2500.0f;              // (i/SN)/2 = i*5000/2
    const float VsnNom = rTbA * asinh_f(halfJ * __builtin_amdgcn_rcpf(Jn0));
    const float VspNom = rTbA * asinh_f(halfJ * __builtin_amdgcn_rcpf(Jp0));

    const float Tb2  = Tb;
    const float Vo2  = fmaf(VoNom  - Vo,  0.1f,    Vo);
    const float Vsn2 = fmaf(VsnNom - Vsn, INV_TSN, Vsn);
    const float Vsp2 = fmaf(VspNom - Vsp, INV_TSN, Vsp);
    const float qnB2 = qnB - qdn;
    const float qnS2 = qnS + qdn - i_;
    const float qpB2 = qpB - qdp;
    const float qpS2 = qpS + i_ + qdp;

    // --- getNextOutput ---
    const float xpo = qpS2 * rq;
    const float xno = qnS2 * rq;

    // MLP_p: 1 -> 8 (tanh) -> 4 (tanh) -> 1
    float h0[8];
#pragma unroll
    for (int j = 0; j < 8; ++j) h0[j] = tanh_f(fmaf(xpo, w0[j], c0[j]));
    float h1[4];
#pragma unroll
    for (int k = 0; k < 4; ++k) {
      float acc = c2[k];
#pragma unroll
      for (int j = 0; j < 8; ++j) acc = fmaf(w2[k * 8 + j], h0[j], acc);
      h1[k] = tanh_f(acc);
    }
    float VepMLP = c4;
#pragma unroll
    for (int k = 0; k < 4; ++k) VepMLP = fmaf(w4[k], h1[k], VepMLP);
    const float VenMLP = fmaf(xno, wnv, cnv);

    const float slp = clamp_f(__fdividef(1.0f - xpo, xpo), 1e-18f, 1e18f);
    const float sln = clamp_f(__fdividef(1.0f - xno, xno), 1e-18f, 1e18f);
    const float rTbF = RF * Tb2;  // R*Tb/F
    const float Vep = 4.03f + rTbF * __logf(slp) + VepMLP;
    const float Ven = 0.01f + rTbF * __logf(sln) + VenMLP;
    const float V   = Vep - Ven - Vo2 - Vsn2 - Vsp2;

    if (idx < n) {
      out[idx] = V;  // V block: d_out[0 .. n)
      float4 o03; o03.x = Tb2;  o03.y = Vo2;  o03.z = Vsn2; o03.w = Vsp2;
      float4 o47; o47.x = qnB2; o47.y = qnS2; o47.z = qpB2; o47.w = qpS2;
      // XNew block: d_out[n + idx*8 ...); n = 4M -> 16B aligned, idx*8 floats = 32B
      float4* xr = reinterpret_cast<float4*>(out + (size_t)n + (size_t)idx * 8);
      xr[0] = o03;
      xr[1] = o47;
    }
  }
}

extern "C" void kernel_launch(void* const* d_in, const int* in_sizes, int n_in,
                              void* d_out, int out_size, void* d_ws, size_t ws_size,
                              hipStream_t stream) {
  const float* inputs = (const float*)d_in[0];
  const float* states = (const float*)d_in[1];
  const float* qMaxP  = (const float*)d_in[2];
  const float* RoP    = (const float*)d_in[3];
  const float* Wp0    = (const float*)d_in[4];
  const float* bp0    = (const float*)d_in[5];
  const float* Wp2    = (const float*)d_in[6];
  const float* bp2    = (const float*)d_in[7];
  const float* Wp4    = (const float*)d_in[8];
  const float* bp4    = (const float*)d_in[9];
  const float* Wn     = (const float*)d_in[10];
  const float* bn     = (const float*)d_in[11];
  float* out = (float*)d_out;

  const int n = in_sizes[0];  // B (inputs is [B,1])
  const int elems_per_block = TPB * EPT;
  const int blocks = (n + elems_per_block - 1) / elems_per_block;

  hipLaunchKernelGGL(battery_cell_kernel, dim3(blocks), dim3(TPB), 0, stream,
                     inputs, states, qMaxP, RoP,
                     Wp0, bp0, Wp2, bp2, Wp4, bp4, Wn, bn,
                     out, n);
}